// TopKClassificationLoss_27187142984006
// MI455X (gfx1250) — compile-verified
//
#include <hip/hip_runtime.h>
#include <stdint.h>

// Soft-rank top-k classification loss for MI455X (gfx1250, wave32).
//
// Roofline: inputs are 256KB total, output is a scalar -> pure latency-bound.
// Strategy: 1 workgroup per row (64 blocks x 256 threads = 8 waves).
//   1. TDM tensor_load_to_lds DMAs the 1000-float row into LDS (CDNA5 path,
//      tracked by TENSORcnt), padded to 1024 with -1e30.
//   2. pos = #elements strictly greater than the target logit (parallel count).
//   3. 1024-wide descending bitonic sort in LDS (55 steps, barrier per step).
//   4. Exact sequential PAV isotonic regression (non-increasing fit of
//      s_i - (N - i)) on lane 0 -> rank at pos -> relu(N - rank) into d_ws.
//   5. Single-wave reduce kernel sums the 64 row losses via
//      v_wmma_f32_16x16x4_f32 (A = ones 16x4, B = losses 4x16; row 0 of D is
//      the column sums) and writes mean to d_out.

#define NCLS 1000
#define NPAD 1024
#define NROWS 64
#define BIGF 1e30f

typedef float v2f  __attribute__((ext_vector_type(2)));
typedef float v8f  __attribute__((ext_vector_type(8)));
typedef unsigned int u32x4 __attribute__((ext_vector_type(4)));
typedef int   i32x4 __attribute__((ext_vector_type(4)));
typedef int   i32x8 __attribute__((ext_vector_type(8)));

__global__ __launch_bounds__(256) void softrank_row_kernel(
    const float* __restrict__ logits, const int* __restrict__ target,
    float* __restrict__ row_loss) {
  __shared__ float s_val[NPAD];
  __shared__ float s_mean[NCLS];
  __shared__ int   s_cnt[NCLS];
  __shared__ int   sh_tgt;
  __shared__ int   sh_gt;

  const int tid = threadIdx.x;
  const int b   = blockIdx.x;
  const float* row = logits + (size_t)b * NCLS;

  if (tid == 0) { sh_tgt = target[b]; sh_gt = 0; }

  // ---- TDM: DMA the row (1000 f32) from global into LDS (wave 0 issues) ----
  if (tid < 32) {
    unsigned long long ga = (unsigned long long)(uintptr_t)row;
    unsigned int lds_off  = (unsigned int)(uintptr_t)(&s_val[0]);
    // D# group 0: count=1, lds_addr, global_addr[56:0], type=2 ("image")
    u32x4 g0;
    g0.x = 1u;
    g0.y = lds_off;
    g0.z = (unsigned int)(ga & 0xFFFFFFFFull);
    g0.w = (unsigned int)((ga >> 32) & 0x01FFFFFFull) | (2u << 30);
    // D# group 1: data_size=4B, tensor_dim0=1000, tensor_dim1=1,
    //             tile_dim0=1000, tile_dim1=1, stride0=1000
    i32x8 g1;
    g1[0] = (int)(2u << 16);               // data_size = 2 -> 4 bytes
    g1[1] = (int)((unsigned)NCLS << 16);   // tensor_dim0[15:0] in bits[63:48]
    g1[2] = (int)(1u << 16);               // tensor_dim0 hi=0, tensor_dim1 lo=1
    g1[3] = (int)((unsigned)NCLS << 16);   // tensor_dim1 hi=0, tile_dim0=1000
    g1[4] = 1;                             // tile_dim1=1, tile_dim2=0
    g1[5] = NCLS;                          // tensor_dim0_stride low 32
    g1[6] = 0;
    g1[7] = 0;
    i32x4 gz = {0, 0, 0, 0};
#if __clang_major__ >= 23
    i32x8 gz8 = {0, 0, 0, 0, 0, 0, 0, 0};
    __builtin_amdgcn_tensor_load_to_lds(g0, g1, gz, gz, gz8, 0);
#else
    __builtin_amdgcn_tensor_load_to_lds(g0, g1, gz, gz, 0);
#endif
    __builtin_amdgcn_s_wait_tensorcnt(0);
  }
  // Pad tail so the 1024-wide descending bitonic net pushes it to the end.
  for (int i = NCLS + tid; i < NPAD; i += 256) s_val[i] = -BIGF;
  __syncthreads();

  // ---- pos = count of elements strictly greater than the target logit ----
  const float tval = s_val[sh_tgt];
  int local = 0;
  for (int i = tid; i < NCLS; i += 256) local += (s_val[i] > tval) ? 1 : 0;
  atomicAdd(&sh_gt, local);
  __syncthreads();
  const int pos = sh_gt;

  // ---- descending bitonic sort of 1024 values, 512 pairs / step ----
  for (int k = 2; k <= NPAD; k <<= 1) {
    for (int j = k >> 1; j > 0; j >>= 1) {
      for (int t = tid; t < (NPAD / 2); t += 256) {
        int i = ((t & ~(j - 1)) << 1) | (t & (j - 1));
        int p = i | j;
        float a = s_val[i];
        float c = s_val[p];
        bool desc = ((i & k) == 0);
        if (desc ? (a < c) : (a > c)) { s_val[i] = c; s_val[p] = a; }
      }
      __syncthreads();
    }
  }

  // ---- exact PAV: non-increasing isotonic fit of v_i = s_i - (N - i) ----
  if (tid == 0) {
    int top = 0;
    for (int i = 0; i < NCLS; ++i) {
      float m = s_val[i] - (float)(NCLS - i);
      float c = 1.0f;
      while (top > 0 && s_mean[top - 1] < m) {  // merge violating blocks
        float cm = s_mean[top - 1];
        float cc = (float)s_cnt[top - 1];
        m = (m * c + cm * cc) / (c + cc);
        c += cc;
        --top;
      }
      s_mean[top] = m;
      s_cnt[top]  = (int)c;
      ++top;
    }
    // y at index pos = mean of the block containing pos
    int cum = 0;
    float y = s_mean[0];
    for (int bl = 0; bl < top; ++bl) {
      cum += s_cnt[bl];
      if (pos < cum) { y = s_mean[bl]; break; }
    }
    float rank = s_val[pos] - y;                  // ranks_sorted[pos]
    row_loss[b] = fmaxf(0.0f, (float)NCLS - rank);
  }
}

// Single-wave reduction: sum 64 losses with one v_wmma_f32_16x16x4_f32.
// A = ones (16x4), B[k][n] = L[16k+n] (4x16) -> D[0][n] = column sums.
__global__ __launch_bounds__(32) void reduce_mean_kernel(
    const float* __restrict__ row_loss, float* __restrict__ out) {
  __shared__ float red[16];
  const int lane = threadIdx.x;

  v2f a;
  a.x = 1.0f;
  a.y = 1.0f;
  // B fragment layout (32-bit B 4x16): lanes 0-15 hold N=lane, VGPR0=K0,
  // VGPR1=K1; lanes 16-31 hold N=lane-16, VGPR0=K2, VGPR1=K3.
  v2f bb;
  if (lane < 16) {
    bb.x = row_loss[lane];
    bb.y = row_loss[lane + 16];
  } else {
    bb.x = row_loss[lane + 16];  // K=2, N=lane-16 -> L[32 + (lane-16)]
    bb.y = row_loss[lane + 32];  // K=3, N=lane-16 -> L[48 + (lane-16)]
  }
  v8f c = {};
  c = __builtin_amdgcn_wmma_f32_16x16x4_f32(
      /*neg_a=*/false, a, /*neg_b=*/false, bb,
      /*c_mod=*/(short)0, c, /*reuse_a=*/false, /*reuse_b=*/false);

  // Row 0 of D lives in VGPR0 of lanes 0..15.
  if (lane < 16) red[lane] = c[0];
  __syncthreads();
  if (lane == 0) {
    float s = 0.0f;
    for (int i = 0; i < 16; ++i) s += red[i];
    out[0] = s * (1.0f / (float)NROWS);
  }
}

extern "C" void kernel_launch(void* const* d_in, const int* in_sizes, int n_in,
                              void* d_out, int out_size, void* d_ws, size_t ws_size,
                              hipStream_t stream) {
  const float* logits = (const float*)d_in[0];
  const int*   target = (const int*)d_in[1];
  float* row_loss = (float*)d_ws;  // 64 floats of scratch

  softrank_row_kernel<<<NROWS, 256, 0, stream>>>(logits, target, row_loss);
  reduce_mean_kernel<<<1, 32, 0, stream>>>(row_loss, (float*)d_out);
}